// time_change_func_M_MGN_26963804684525
// MI455X (gfx1250) — compile-verified
//
#include <hip/hip_runtime.h>
#include <hip/hip_bf16.h>

typedef float v2f __attribute__((ext_vector_type(2)));
typedef float v8f __attribute__((ext_vector_type(8)));

#define K_UNITS 64
#define N_DIM   128
#define T_PTS   32768
#define EPS_F   0.01f

__device__ __forceinline__ float bperm_f(int src_lane, float v) {
    return __int_as_float(__builtin_amdgcn_ds_bpermute(src_lane << 2, __float_as_int(v)));
}

// ---------------- Kernel A: init min-key, compute sum(V*V), prefetch W/b ----
__global__ void mgn_init_kernel(const float* __restrict__ V,
                                const float* __restrict__ W,
                                const float* __restrict__ B,
                                float* __restrict__ wsF,
                                unsigned* __restrict__ wsU) {
    __shared__ float red[128];
    int i = threadIdx.x;                 // 128 threads
    float v = V[i];
    red[i] = v * v;
    __syncthreads();
    for (int s = 64; s > 0; s >>= 1) {
        if (i < s) red[i] += red[i + s];
        __syncthreads();
    }
    if (i == 0) {
        wsF[1] = red[0];                 // sum(V*V)
        wsU[0] = 0xFFFFFFFFu;            // +inf in monotone-uint key space
    }
    // warm L2 for the main kernel's LDS staging (global_prefetch_b8 path)
    __builtin_prefetch(W + i * 64, 0, 0);
    __builtin_prefetch(B + i * 64, 0, 0);
}

// ---------------- Kernel B: main WMMA compute ------------------------------
// One wave (32 lanes) handles 16 time points. 256 threads = 8 waves/block.
__global__ __launch_bounds__(256) void mgn_main_kernel(
        const float* __restrict__ t,
        const float* __restrict__ W,
        const float* __restrict__ B,
        const float* __restrict__ a,
        float* __restrict__ out,
        const float* __restrict__ wsF,
        unsigned* __restrict__ wsU) {
    __shared__ float lW[K_UNITS * N_DIM];   // 32 KB
    __shared__ float lB[K_UNITS * N_DIM];   // 32 KB

    const int tid = threadIdx.x;
    // Stage W and b into LDS with float4 loads (fits in L2 anyway; LDS removes latency)
    {
        const float4* W4 = (const float4*)W;
        const float4* B4 = (const float4*)B;
        float4* lW4 = (float4*)lW;
        float4* lB4 = (float4*)lB;
        for (int i = tid; i < (K_UNITS * N_DIM) / 4; i += 256) {
            lW4[i] = W4[i];
            lB4[i] = B4[i];
        }
    }
    __syncthreads();

    const int lane  = tid & 31;
    const int wave  = tid >> 5;
    const int tile  = blockIdx.x * 8 + wave;      // 256 blocks * 8 waves = 2048 tiles of 16 t
    const int j     = lane & 15;                  // time index within tile
    const int half2 = (lane >> 4) << 1;           // lanes 0-15 -> n+{0,1}; lanes 16-31 -> n+{2,3}

    const float tl  = t[tile * 16 + j];
    const float sVV = wsF[1];
    const float a0s = a[0];

    float nn = 0.0f;

    for (int k = 0; k < K_UNITS; ++k) {
        v8f c1 = {0.f,0.f,0.f,0.f,0.f,0.f,0.f,0.f};   // Gram accumulator (diag = sum r^2)
        v8f c2 = {0.f,0.f,0.f,0.f,0.f,0.f,0.f,0.f};   // gate accumulator (row0 = sum W*r)
        const int base = k * N_DIM + half2;
        #pragma unroll 4
        for (int c = 0; c < N_DIM / 4; ++c) {
            const int idx = base + c * 4;
            const float w0 = lW[idx], w1 = lW[idx + 1];
            const float b0 = lB[idx], b1 = lB[idx + 1];
            v2f av, wv;
            av.x = fmaxf(fmaf(w0, tl, b0), 0.0f);     // relu(W*t+b)   (VALU)
            av.y = fmaxf(fmaf(w1, tl, b1), 0.0f);
            wv.x = w0; wv.y = w1;
            // D[i,j] += sum_n r[n,ti]*r[n,tj]  (A-as-B == transpose for 16x16x4 f32 layout)
            c1 = __builtin_amdgcn_wmma_f32_16x16x4_f32(false, av, false, av, (short)0, c1, false, false);
            // D[i,j] += sum_n W[n]*r[n,tj]     (every row identical = gate)
            c2 = __builtin_amdgcn_wmma_f32_16x16x4_f32(false, wv, false, av, (short)0, c2, false, false);
        }
        // Extract diag(c1): diag[j] at lane j (VGPR j) for j<8, at lane 16+j (VGPR j-8) for j>=8.
        const int srcl = (j < 8) ? j : (16 + j);
        const float s0 = bperm_f(srcl, c1[0]);
        const float s1 = bperm_f(srcl, c1[1]);
        const float s2 = bperm_f(srcl, c1[2]);
        const float s3 = bperm_f(srcl, c1[3]);
        const float s4 = bperm_f(srcl, c1[4]);
        const float s5 = bperm_f(srcl, c1[5]);
        const float s6 = bperm_f(srcl, c1[6]);
        const float s7 = bperm_f(srcl, c1[7]);
        const int g = j & 7;
        const float p0 = (g & 1) ? s1 : s0;
        const float p1 = (g & 1) ? s3 : s2;
        const float p2 = (g & 1) ? s5 : s4;
        const float p3 = (g & 1) ? s7 : s6;
        const float q0 = (g & 2) ? p1 : p0;
        const float q1 = (g & 2) ? p3 : p2;
        const float diag = (g & 4) ? q1 : q0;       // sum_n r^2 for this lane's t
        const float gate = c2[0];                   // valid on all 32 lanes (rows identical)
        nn = fmaf(0.5f * diag, gate, nn);           // prim_k * gate_k
    }

    const float mgn = fmaf(sVV, tl, a0s) + nn;
    if (lane < 16) out[tile * 16 + j] = mgn;

    // Wave min (lanes 16-31 are duplicates; min unaffected), then one atomic per wave.
    float m = mgn;
    #pragma unroll
    for (int off = 16; off > 0; off >>= 1)
        m = fminf(m, bperm_f(lane ^ off, m));
    if (lane == 0) {
        const unsigned u = __float_as_uint(m);
        const unsigned key = (u & 0x80000000u) ? ~u : (u | 0x80000000u); // monotone map
        atomicMin(wsU, key);
    }
}

// ---------------- Kernel C: apply positif_phi_t correction -----------------
__global__ void mgn_fix_kernel(float* __restrict__ out, const unsigned* __restrict__ wsU) {
    const unsigned key = wsU[0];
    const float mn = (key & 0x80000000u) ? __uint_as_float(key & 0x7FFFFFFFu)
                                         : __uint_as_float(~key);
    const float corr = (mn <= 0.0f) ? (EPS_F - mn) : 0.0f;
    const int tt = blockIdx.x * blockDim.x + threadIdx.x;
    out[tt] += corr;
}

extern "C" void kernel_launch(void* const* d_in, const int* in_sizes, int n_in,
                              void* d_out, int out_size, void* d_ws, size_t ws_size,
                              hipStream_t stream) {
    const float* t = (const float*)d_in[0];   // [T]
    const float* W = (const float*)d_in[1];   // [K,N]
    const float* B = (const float*)d_in[2];   // [K,N]
    const float* V = (const float*)d_in[3];   // [N,1]
    const float* a = (const float*)d_in[4];   // [1,1]
    float* out     = (float*)d_out;           // [T]
    float* wsF     = (float*)d_ws;            // wsF[1] = sum(V*V)
    unsigned* wsU  = (unsigned*)d_ws;         // wsU[0] = min key
    (void)in_sizes; (void)n_in; (void)out_size; (void)ws_size;

    mgn_init_kernel<<<1, 128, 0, stream>>>(V, W, B, wsF, wsU);
    mgn_main_kernel<<<T_PTS / (8 * 16), 256, 0, stream>>>(t, W, B, a, out, wsF, wsU);
    mgn_fix_kernel<<<T_PTS / 256, 256, 0, stream>>>(out, wsU);
}